// Wavelet_55911884259710
// MI455X (gfx1250) — compile-verified
//
#include <hip/hip_runtime.h>

#ifndef __has_builtin
#define __has_builtin(x) 0
#endif

#define NTHREADS 256
#define CHUNK    4096
#define HX       64                 // x-halo (circular); generous vs required ~41
#define NX       (CHUNK + HX)       // 4160 floats of x per tile
#define NH       (NX / 2)           // 2080 half-grid slots
#define NQ       (NH / 2)           // 1040 quarter-grid slots

#define L_LEN 131072
#define BATCH 128
#define HALF0 (L_LEN / 2)           // 65536
#define QUAR0 (L_LEN / 4)           // 32768

typedef __attribute__((address_space(1))) float gfloat;
typedef __attribute__((address_space(3))) float lfloat;

// Async-to-LDS builtin wants pointers to 16-byte int vectors in AS1/AS3
// (diagnostic: "__attribute__((__vector_size__(4 * sizeof(int)))) int __device__ *").
typedef int v4i __attribute__((vector_size(16)));
typedef __attribute__((address_space(1))) v4i gv4i;
typedef __attribute__((address_space(3))) v4i lv4i;

#if __has_builtin(__builtin_amdgcn_global_load_async_to_lds_b128)
#define HAVE_ASYNC_B128 1
#else
#define HAVE_ASYNC_B128 0
#endif
#if __has_builtin(__builtin_amdgcn_s_wait_asynccnt)
#define HAVE_WAIT_ASYNC 1
#else
#define HAVE_WAIT_ASYNC 0
#endif

// Fused 2-level lifting wavelet. One block = one 4096-col chunk of one row.
// conv(a,b)[i] = b0*a[i] + b1*a[i+1] + b2*a[i+2]  (forward taps, circular).
__global__ __launch_bounds__(NTHREADS)
void wavelet2lvl_fused(const float* __restrict__ x,
                       const float* __restrict__ Pc,   // (2,2,3)
                       const float* __restrict__ Uc,   // (2,2,3)
                       float* __restrict__ out0,       // avg level0 (128,65536)
                       float* __restrict__ out1,       // avg level1 (128,32768)
                       float* __restrict__ out2,       // det level1 (128,32768)
                       float* __restrict__ outS)       // sum_log_dets scalar
{
    __shared__ __attribute__((aligned(16))) float sx[NX];   // x tile
    __shared__ float sd0[NH];   // level0 k=0 detail
    __shared__ float sa0[NH];   // level0 k=0 average
    __shared__ float sd1[NH];   // level0 k=1 detail  (= det0, feeds level1)
    __shared__ float sq0[NQ];   // level1 k=0 detail
    __shared__ float sq1[NQ];   // level1 k=0 average
    __shared__ float sq2[NQ];   // level1 k=1 detail  (= det1)

    const int tid  = threadIdx.x;
    const int row  = blockIdx.y;
    const int cb   = blockIdx.x;
    const int base = cb * CHUNK;                       // x column base of chunk
    const float* xrow = x + (size_t)row * L_LEN;

    // ---------- stage x tile (+ circular halo) into LDS ----------
    // base and L_LEN are multiples of 4 floats, so each float4 piece either
    // lies fully before the row end or wraps as a whole: wrap = col - L.
#if HAVE_ASYNC_B128
    {
        gfloat* xg = (gfloat*)xrow;
        lfloat* sl = (lfloat*)sx;
        for (int v = tid; v < NX / 4; v += NTHREADS) {
            int col = base + v * 4;
            if (col >= L_LEN) col -= L_LEN;
            __builtin_amdgcn_global_load_async_to_lds_b128(
                (gv4i*)(xg + col), (lv4i*)(sl + v * 4), 0, 0);
        }
    }
#if HAVE_WAIT_ASYNC
    __builtin_amdgcn_s_wait_asynccnt(0);
#else
    asm volatile("s_wait_asynccnt 0x0" ::: "memory");
#endif
#else
    for (int v = tid; v < NX / 4; v += NTHREADS) {
        int col = base + v * 4;
        if (col >= L_LEN) col -= L_LEN;
        *(float4*)(sx + v * 4) = *(const float4*)(xrow + col);
    }
#endif
    __syncthreads();

    // Coefficients: P[level][k][tap], stride 6 per level, 3 per k. Uniform
    // kernarg-relative addresses -> scalar loads, L2-resident.
    const float p00a = Pc[0], p00b = Pc[1],  p00c = Pc[2];
    const float p01a = Pc[3], p01b = Pc[4],  p01c = Pc[5];
    const float p10a = Pc[6], p10b = Pc[7],  p10c = Pc[8];
    const float p11a = Pc[9], p11b = Pc[10], p11c = Pc[11];
    const float u00a = Uc[0], u00b = Uc[1],  u00c = Uc[2];
    const float u01a = Uc[3], u01b = Uc[4],  u01c = Uc[5];
    const float u10a = Uc[6], u10b = Uc[7],  u10c = Uc[8];
    const float u11a = Uc[9], u11b = Uc[10], u11c = Uc[11];

    // ---------- level 0, lifting k=0: d0 = odd - conv(even, P00) ----------
    for (int h = tid; h < NH - 2; h += NTHREADS) {
        float e0 = sx[2*h], e1 = sx[2*h + 2], e2 = sx[2*h + 4];
        sd0[h] = sx[2*h + 1] - (p00a*e0 + p00b*e1 + p00c*e2);
    }
    __syncthreads();

    // a0 = even + conv(d0, U00)
    for (int h = tid; h < NH - 4; h += NTHREADS)
        sa0[h] = sx[2*h] + (u00a*sd0[h] + u00b*sd0[h+1] + u00c*sd0[h+2]);
    __syncthreads();

    // ---------- level 0, lifting k=1: d1 = d0 - conv(a0, P01) ----------
    for (int h = tid; h < NH - 6; h += NTHREADS)
        sd1[h] = sd0[h] - (p01a*sa0[h] + p01b*sa0[h+1] + p01c*sa0[h+2]);
    __syncthreads();

    // a1 = a0 + conv(d1, U01)  -> zs[0]; and level1 k=0 detail from det0=sd1
    {
        const size_t ob0 = (size_t)row * HALF0 + (size_t)cb * (CHUNK / 2);
        for (int h = tid; h < CHUNK / 2; h += NTHREADS)
            out0[ob0 + h] = sa0[h] + (u01a*sd1[h] + u01b*sd1[h+1] + u01c*sd1[h+2]);
    }
    for (int q = tid; q < NQ - 5; q += NTHREADS) {      // needs sd1[2q+5] < NH-6
        float e0 = sd1[2*q], e1 = sd1[2*q + 2], e2 = sd1[2*q + 4];
        sq0[q] = sd1[2*q + 1] - (p10a*e0 + p10b*e1 + p10c*e2);
    }
    __syncthreads();

    // level1 k=0 average: aa0 = e1 + conv(dd0, U10)
    for (int q = tid; q < NQ - 7; q += NTHREADS)
        sq1[q] = sd1[2*q] + (u10a*sq0[q] + u10b*sq0[q+1] + u10c*sq0[q+2]);
    __syncthreads();

    // level1 k=1 detail: dd1 = dd0 - conv(aa0, P11)  -> zs[2]
    const size_t ob1 = (size_t)row * QUAR0 + (size_t)cb * (CHUNK / 4);
    for (int q = tid; q < NQ - 9; q += NTHREADS) {
        float dd1 = sq0[q] - (p11a*sq1[q] + p11b*sq1[q+1] + p11c*sq1[q+2]);
        sq2[q] = dd1;
        if (q < CHUNK / 4) out2[ob1 + q] = dd1;
    }
    __syncthreads();

    // level1 k=1 average: aa1 = aa0 + conv(dd1, U11)  -> zs[1]
    for (int q = tid; q < CHUNK / 4; q += NTHREADS)
        out1[ob1 + q] = sq1[q] + (u11a*sq2[q] + u11b*sq2[q+1] + u11c*sq2[q+2]);

    // sum_log_dets == 0.0 scalar (last output element)
    if (tid == 0 && cb == 0 && row == 0) outS[0] = 0.0f;
}

extern "C" void kernel_launch(void* const* d_in, const int* in_sizes, int n_in,
                              void* d_out, int out_size, void* d_ws, size_t ws_size,
                              hipStream_t stream) {
    (void)in_sizes; (void)n_in; (void)d_ws; (void)ws_size;
    const float* x  = (const float*)d_in[0];
    const float* Pc = (const float*)d_in[1];
    const float* Uc = (const float*)d_in[2];

    float* out  = (float*)d_out;
    float* out0 = out;                                   // (128, 65536)
    float* out1 = out0 + (size_t)BATCH * HALF0;          // (128, 32768)
    float* out2 = out1 + (size_t)BATCH * QUAR0;          // (128, 32768)
    float* outS = out + ((size_t)out_size - 1);          // scalar

    dim3 grid(L_LEN / CHUNK, BATCH);                     // 32 x 128 blocks
    wavelet2lvl_fused<<<grid, NTHREADS, 0, stream>>>(x, Pc, Uc, out0, out1, out2, outS);
}